// Transformer2D_47622597378151
// MI455X (gfx1250) — compile-verified
//
#include <hip/hip_runtime.h>

// Flow-warp bilinear sampling (grid_sample, border padding): sample src at
// pixel coords (grid + flow), clamped to the image rect.
// src: [B,C,H,W] f32, flow: [B,2,H,W] f32 (ch0 = dy, ch1 = dx), out: [B,C,H,W] f32.
//
// Roofline: ~264 MB min traffic -> ~11 us at 23.3 TB/s; src (128 MB) fits in
// the 192 MB L2 so bilinear tap overlap is cache-absorbed. One thread per
// pixel, weights computed once, 32 channels walked with the 1 MiB channel
// stride folded into the 24-bit signed instruction offset (8-channel unroll),
// B64 row-pair loads, global_prefetch_b8 warming the next channel chunk, and
// the flow field staged through LDS with the gfx1250 async-to-LDS engine
// (global_load_async_to_lds_b32 + s_wait_asynccnt).

namespace {

constexpr int B = 4, C = 32, H = 512, W = 512;
constexpr int HW = H * W;                       // 262144 = 2^18
constexpr unsigned CH_BYTES = (unsigned)HW * 4; // 1 MiB per-channel stride

// 8-byte value with only 4-byte alignment guarantee (xb may be odd).
struct __attribute__((aligned(4))) F2 { float x, y; };

__device__ __forceinline__ F2 load_f2(const char* p) {
  return *reinterpret_cast<const F2*>(p);
}

} // namespace

__global__ __launch_bounds__(256) void flow_warp_bilinear_kernel(
    const float* __restrict__ src, const float* __restrict__ flow,
    float* __restrict__ out) {
  const unsigned tid = blockIdx.x * 256u + threadIdx.x; // 0 .. B*H*W-1
  const unsigned b   = tid >> 18;          // HW = 2^18
  const unsigned pix = tid & (HW - 1);
  const unsigned y   = pix >> 9;           // W = 2^9
  const unsigned x   = pix & (W - 1);

  // ---- Stage flow (dy, dx) for this block's 256 pixels through LDS using the
  // gfx1250 async copy engine. Wave-local: each lane copies its own element to
  // its own slot, so s_wait_asynccnt 0 is the only sync needed (no barrier).
  // The dx plane sits one channel stride (1 MiB) away -> 24-bit offset imm.
  __shared__ float sflow[2 * 256];
  {
    // Flat->LDS aperture maps addr[31:0] directly to the LDS byte offset.
    const unsigned lds_dy =
        (unsigned)(uintptr_t)(const void*)&sflow[threadIdx.x];
    const unsigned lds_dx = lds_dy + 256u * 4u;
    const float* gdy = flow + ((b * 2u + 0u) * (unsigned)HW + pix);
    asm volatile(
        "global_load_async_to_lds_b32 %0, %2, off\n\t"
        "global_load_async_to_lds_b32 %1, %2, off offset:1048576\n\t"
        "s_wait_asynccnt 0x0"
        :
        : "v"(lds_dy), "v"(lds_dx), "v"(gdy)
        : "memory");
  }
  const float dy = sflow[threadIdx.x];         // ds_load_b32
  const float dx = sflow[threadIdx.x + 256];   // ds_load_b32

  // Border padding == clamp sample coords into [0, dim-1].
  const float fy = fminf(fmaxf((float)y + dy, 0.0f), (float)(H - 1));
  const float fx = fminf(fmaxf((float)x + dx, 0.0f), (float)(W - 1));

  const float y0f = floorf(fy);
  const float x0f = floorf(fx);
  const float wy = fy - y0f;
  const float wx = fx - x0f;
  const int iy0 = (int)y0f;
  const int ix0 = (int)x0f;
  const int iy1 = min(iy0 + 1, H - 1);
  const int ixb = min(ix0, W - 2);     // pair base so [ixb, ixb+1] is in-row
  const bool selx = (ix0 > ixb);       // x0 hit the right border (then wx == 0)

  // Fold the border select into the x-weights once (no per-channel cndmask):
  //   normal: (1-wx) applies to pair.x (== v(x0)), wx to pair.y (== v(x1))
  //   border: v00 = v01 = pair.y, wx = 0 -> weights (0, 1).
  const float axw = selx ? 0.0f : (1.0f - wx);
  const float bxw = selx ? 1.0f : wx;
  const float omwy = 1.0f - wy;
  const float c00 = omwy * axw;
  const float c01 = omwy * bxw;
  const float c10 = wy * axw;
  const float c11 = wy * bxw;

  const char* sp = (const char*)src;
  char*       op = (char*)out;

  // Byte offsets for channel 0; everything fits comfortably in 32 bits (128 MB).
  unsigned off0 = (unsigned)(((b * (unsigned)(C * H) + (unsigned)iy0) * (unsigned)W
                              + (unsigned)ixb) * 4u);
  unsigned off1 = (unsigned)(((b * (unsigned)(C * H) + (unsigned)iy1) * (unsigned)W
                              + (unsigned)ixb) * 4u);
  unsigned offo = (b * (unsigned)(C * HW) + pix) * 4u;

  // 4 chunks of 8 channels: intra-chunk channel strides (0..7 MiB) ride in the
  // signed 24-bit instruction offset; 16 B64 loads in flight per chunk.
#pragma unroll 1
  for (int cc = 0; cc < C; cc += 8) {
    if (cc + 8 < C) {
      // Warm L2 for the next channel chunk (emits global_prefetch_b8).
      __builtin_prefetch(sp + (off0 + 8u * CH_BYTES), 0, 3);
      __builtin_prefetch(sp + (off1 + 8u * CH_BYTES), 0, 3);
    }
#pragma unroll
    for (int c = 0; c < 8; ++c) {
      const unsigned cb = (unsigned)c * CH_BYTES;
      const F2 r0 = load_f2(sp + (off0 + cb));   // row y0: v(x0-ish), v(x1)
      const F2 r1 = load_f2(sp + (off1 + cb));   // row y1
      const float o = fmaf(r1.y, c11,
                      fmaf(r1.x, c10,
                      fmaf(r0.y, c01, r0.x * c00)));
      *(float*)(op + (offo + cb)) = o;
    }
    off0 += 8u * CH_BYTES;
    off1 += 8u * CH_BYTES;
    offo += 8u * CH_BYTES;
  }
}

extern "C" void kernel_launch(void* const* d_in, const int* in_sizes, int n_in,
                              void* d_out, int out_size, void* d_ws, size_t ws_size,
                              hipStream_t stream) {
  (void)in_sizes; (void)n_in; (void)out_size; (void)d_ws; (void)ws_size;
  const float* src  = (const float*)d_in[0];
  const float* flow = (const float*)d_in[1];
  float* out = (float*)d_out;

  const int threads = 256;
  const int blocks  = (B * H * W) / threads;   // 4096
  flow_warp_bilinear_kernel<<<blocks, threads, 0, stream>>>(src, flow, out);
}